// MoD_16999480557997
// MI455X (gfx1250) — compile-verified
//
#include <hip/hip_runtime.h>
#include <hip/hip_bf16.h>

// MoD (Mixture-of-Depths) routing block, B=4, S=4096, D=2048, capacity 0.5.
// out[b,s,:] = x[b,s,:] * (1 + softmax_topk_weight[b,s])
//
// Pass 1: router GEMV via V_WMMA_F32_16X16X4_F32 (reads x, warms 192MB L2)
// Pass 2: per-batch exact top-k threshold (radix select) + softmax -> w_tok
// Pass 3: bandwidth pass, x from L2, non-temporal stores for out

#define MOD_B 4
#define MOD_S 4096
#define MOD_D 2048
#define MOD_K 2048  // S * 0.5

typedef __attribute__((ext_vector_type(2))) float v2f;
typedef __attribute__((ext_vector_type(4))) float v4f;
typedef __attribute__((ext_vector_type(8))) float v8f;

// ---------------------------------------------------------------------------
// Pass 1: logits[b*S+s] = dot(x[b,s,:], w[:])  via WMMA f32 16x16x4.
// One wave per 16-token tile. A-frag layout (ISA 7.12.2, 32-bit A 16x4):
//   lanes 0-15 : M=lane,  VGPR0=K0, VGPR1=K1
//   lanes 16-31: M=lane-16, VGPR0=K2, VGPR1=K3
// B-frag (4x16) mirrors this; we broadcast w[k..k+3] into all 16 columns, so
// every column of the 16x16 f32 accumulator equals the 16 logits.
// ---------------------------------------------------------------------------
__global__ __launch_bounds__(128) void mod_logits_wmma(
    const float* __restrict__ x, const float* __restrict__ w,
    float* __restrict__ logits) {
  const int lane = threadIdx.x & 31;
  const int wave = threadIdx.x >> 5;
  const int tile = blockIdx.x * (blockDim.x >> 5) + wave;  // 16 tokens / tile
  const int t0 = tile << 4;
  const int row = lane & 15;
  const int koff = (lane >> 4) << 1;  // 0 for lanes 0-15, 2 for lanes 16-31

  const float* pa = x + (size_t)(t0 + row) * MOD_D + koff;
  const float* pb = w + koff;

  v8f c = {};
#pragma unroll 8
  for (int k = 0; k < MOD_D; k += 4) {
    v2f a = *(const v2f*)(pa + k);  // {x[row,k+koff], x[row,k+koff+1]}
    v2f b = *(const v2f*)(pb + k);  // {w[k+koff],     w[k+koff+1]}
    c = __builtin_amdgcn_wmma_f32_16x16x4_f32(
        /*neg_a=*/false, a, /*neg_b=*/false, b,
        /*c_mod=*/(short)0, c, /*reuse_a=*/false, /*reuse_b=*/false);
  }

  // C/D layout: VGPR r, lanes 0-15 -> (M=r, N=lane); lanes 16-31 -> (M=8+r).
  // Column N=0 lives in lanes 0 and 16.
  if (lane == 0) {
#pragma unroll
    for (int r = 0; r < 8; ++r) logits[t0 + r] = c[r];
  } else if (lane == 16) {
#pragma unroll
    for (int r = 0; r < 8; ++r) logits[t0 + 8 + r] = c[r];
  }
}

// ---------------------------------------------------------------------------
// Pass 2: one block per batch. Exact k-th largest via radix select on
// order-preserving uint keys, jax.lax.top_k tie semantics (earliest index
// wins among equal values), softmax over the selected k logits.
// Writes w_tok[b*S+s] (0 for unselected tokens).
// ---------------------------------------------------------------------------
__device__ __forceinline__ unsigned f2key(float f) {
  unsigned u = __float_as_uint(f);
  return (u >> 31) ? ~u : (u | 0x80000000u);
}
__device__ __forceinline__ float key2f(unsigned k) {
  unsigned u = (k & 0x80000000u) ? (k & 0x7FFFFFFFu) : ~k;
  return __uint_as_float(u);
}

__global__ __launch_bounds__(256) void mod_select(
    const float* __restrict__ logits, float* __restrict__ wtok) {
  __shared__ unsigned keys[MOD_S];
  __shared__ unsigned cnt;
  __shared__ unsigned maxkey;
  __shared__ float ssum;

  const int b = blockIdx.x;
  const int tid = threadIdx.x;
  const float* lg = logits + b * MOD_S;

  for (int i = tid; i < MOD_S; i += 256) keys[i] = f2key(lg[i]);
  __syncthreads();

  // Radix select: find the MOD_K-th largest key.
  unsigned prefix = 0;
  int remaining = MOD_K;
  for (int bit = 31; bit >= 0; --bit) {
    const unsigned maskHigh = (bit == 31) ? 0u : (0xFFFFFFFFu << (bit + 1));
    const unsigned bmask = 1u << bit;
    if (tid == 0) cnt = 0;
    __syncthreads();
    unsigned local = 0;
    for (int i = tid; i < MOD_S; i += 256) {
      unsigned kk = keys[i];
      if (((kk ^ prefix) & maskHigh) == 0 && (kk & bmask)) local++;
    }
    if (local) atomicAdd(&cnt, local);
    __syncthreads();
    unsigned c1 = cnt;
    if ((int)c1 >= remaining) {
      prefix |= bmask;
    } else {
      remaining -= (int)c1;
    }
    __syncthreads();  // everyone read cnt before next reset
  }
  const unsigned Tkey = prefix;     // k-th largest key
  const int need = remaining;       // ties at Tkey to take (index order)
  const float Tval = key2f(Tkey);

  // Max logit (top-1 is always selected, so global max == selected max).
  if (tid == 0) { maxkey = 0; ssum = 0.0f; }
  __syncthreads();
  unsigned lm = 0;
  for (int i = tid; i < MOD_S; i += 256) lm = max(lm, keys[i]);
  atomicMax(&maxkey, lm);
  __syncthreads();
  const float mv = key2f(maxkey);

  // Softmax denominator over the selected set.
  float lsum = 0.0f;
  for (int i = tid; i < MOD_S; i += 256) {
    if (keys[i] > Tkey) lsum += __expf(lg[i] - mv);
  }
  if (tid == 0) lsum += (float)need * __expf(Tval - mv);
  if (lsum != 0.0f) atomicAdd(&ssum, lsum);
  __syncthreads();
  const float inv = 1.0f / ssum;

  // Weights: strictly-greater tokens are selected; ties handled below.
  for (int i = tid; i < MOD_S; i += 256) {
    float wv = (keys[i] > Tkey) ? __expf(lg[i] - mv) * inv : 0.0f;
    wtok[b * MOD_S + i] = wv;
  }
  __syncthreads();
  if (tid == 0) {
    const float twv = __expf(Tval - mv) * inv;
    int taken = 0;
    for (int s = 0; s < MOD_S && taken < need; ++s) {
      if (keys[s] == Tkey) {
        wtok[b * MOD_S + s] = twv;
        ++taken;
      }
    }
  }
}

// ---------------------------------------------------------------------------
// Pass 3: out = x * (1 + w_tok[token]).  x reads hit L2 (warmed by pass 1);
// out goes through non-temporal stores so the write stream does not evict x.
// ---------------------------------------------------------------------------
__global__ __launch_bounds__(256) void mod_scale(
    const v4f* __restrict__ x4, const float* __restrict__ wtok,
    v4f* __restrict__ out4) {
  const long i = (long)blockIdx.x * blockDim.x + threadIdx.x;
  const long token = i >> 9;  // (i*4) / 2048
  const float s = 1.0f + wtok[token];
  v4f v = x4[i];
  v *= s;
  __builtin_nontemporal_store(v, &out4[i]);
}

// ---------------------------------------------------------------------------
extern "C" void kernel_launch(void* const* d_in, const int* in_sizes, int n_in,
                              void* d_out, int out_size, void* d_ws,
                              size_t ws_size, hipStream_t stream) {
  const float* x = (const float*)d_in[0];        // [B,S,D] f32
  const float* w = (const float*)d_in[1];        // [D]     f32
  float* out = (float*)d_out;                    // [B,S,D] f32

  float* logits = (float*)d_ws;                  // B*S floats
  float* wtok = logits + MOD_B * MOD_S;          // B*S floats

  // Pass 1: 1024 token-tiles of 16, 4 waves (128 thr) per block -> 256 blocks.
  mod_logits_wmma<<<(MOD_B * MOD_S / 16) / 4, 128, 0, stream>>>(x, w, logits);

  // Pass 2: one block per batch row.
  mod_select<<<MOD_B, 256, 0, stream>>>(logits, wtok);

  // Pass 3: B*S*D/4 float4 elements, 256 threads/block.
  const long n4 = (long)MOD_B * MOD_S * MOD_D / 4;
  mod_scale<<<(unsigned)(n4 / 256), 256, 0, stream>>>((const v4f*)x, wtok,
                                                      (v4f*)out);
}